// BandMatrix_54494545052119
// MI455X (gfx1250) — compile-verified
//
#include <hip/hip_runtime.h>
#include <hip/hip_bf16.h>

// Banded GEMM: out[M, N_OUT] = x[M, N_IN] @ W^T + bias, W banded (half-width BW)
// W stored as contiguous per-row band values (COO order), offsets closed-form.
//
// One wave computes a 64(M) x 32(N) tile: 8 f32 WMMA accumulators, sharing
// 4 A-fragments across two N-tiles -> 1.0 vmem per v_wmma_f32_16x16x4_f32.

typedef float v2f __attribute__((ext_vector_type(2)));
typedef float v8f __attribute__((ext_vector_type(8)));

#define N_IN    4096
#define N_OUT   4096
#define BW      512
#define M_TOTAL 8192   // 4 * 2048

__device__ __forceinline__ int band_lo(int o) {
    int l = o - BW;
    return l < 0 ? 0 : l;
}
__device__ __forceinline__ int band_hi(int o) {
    int h = o + BW + 1;
    return h > N_IN ? N_IN : h;
}
// Offset of row o's first value in the packed band-value stream.
__device__ __forceinline__ int band_start(int o) {
    // Region 1: o <= BW           -> count(i) = i + BW + 1
    // Region 2: BW < o <= N_IN-BW -> count = 2*BW+1
    // Region 3: o > N_IN-BW       -> count(N_IN-BW + t) = 2*BW - t
    if (o <= BW)
        return o * (BW + 1) + ((o * (o - 1)) >> 1);
    const int sBW = BW * (BW + 1) + ((BW * (BW - 1)) >> 1);      // start(BW)
    const int r3  = N_IN - BW;
    if (o <= r3)
        return sBW + (o - BW) * (2 * BW + 1);
    const int sr3 = sBW + (r3 - BW) * (2 * BW + 1);              // start(r3)
    const int j   = o - r3;
    return sr3 + 2 * BW * j - ((j * (j - 1)) >> 1);
}

// Branchless edge B-fragment: unconditional clamped loads + cndmask selects.
__device__ __forceinline__ v2f load_b_edge(const float* __restrict__ wrow,
                                           int k, int lo, int hi) {
    int c0 = k;     if (c0 < lo) c0 = lo;  if (c0 > hi - 1) c0 = hi - 1;
    int c1 = k + 1; if (c1 < lo) c1 = lo;  if (c1 > hi - 1) c1 = hi - 1;
    float v0 = wrow[c0];
    float v1 = wrow[c1];
    v2f b;
    b.x = (k     >= lo && k     < hi) ? v0 : 0.0f;
    b.y = (k + 1 >= lo && k + 1 < hi) ? v1 : 0.0f;
    return b;
}

#define WMMA4(bfrag, A0, A1, A2, A3, C0, C1, C2, C3)                                                   \
    C0 = __builtin_amdgcn_wmma_f32_16x16x4_f32(false, A0, false, (bfrag), (short)0, C0, false, false); \
    C1 = __builtin_amdgcn_wmma_f32_16x16x4_f32(false, A1, false, (bfrag), (short)0, C1, false, false); \
    C2 = __builtin_amdgcn_wmma_f32_16x16x4_f32(false, A2, false, (bfrag), (short)0, C2, false, false); \
    C3 = __builtin_amdgcn_wmma_f32_16x16x4_f32(false, A3, false, (bfrag), (short)0, C3, false, false)

__global__ __launch_bounds__(256)
void band_wmma_f32_kernel(const float* __restrict__ x,
                          const float* __restrict__ Wv,
                          const float* __restrict__ bias,
                          float* __restrict__ out)
{
    const int lane = threadIdx.x & 31;
    const int wave = threadIdx.x >> 5;

    const int n0    = blockIdx.x * 32;                 // two 16-wide N tiles
    const int mBase = (blockIdx.y * 8 + wave) * 64;    // 64 M-rows per wave

    // WMMA f32 16x16x4 fragment mapping (wave32):
    //   A (16x4, MxK): lanes 0-15 row=lane, VGPR0=K0,VGPR1=K1; lanes 16-31 row=lane-16, K2/K3
    //   B (4x16, KxN): symmetric — lane&15 selects N-column, lane>>4 selects K half
    //   C/D (16x16):   VGPR v: lanes 0-15 -> (M=v, N=lane), lanes 16-31 -> (M=v+8, N=lane-16)
    const int col   = lane & 15;
    const int khalf = (lane >> 4) << 1;    // 0 or 2

    const int oA = n0 + col;               // N-tile A column
    const int oB = n0 + 16 + col;          // N-tile B column
    const int loA = band_lo(oA), hiA = band_hi(oA);
    const int loB = band_lo(oB), hiB = band_hi(oB);
    const float* wrowA = Wv + band_start(oA) - loA;   // index with absolute k
    const float* wrowB = Wv + band_start(oB) - loB;

    // Wave-uniform K range: union of both tiles' bands.
    int kStart = n0 - BW;           if (kStart < 0)  kStart = 0;
    int kEnd   = n0 + 31 + BW + 1;  if (kEnd > N_IN) kEnd = N_IN;

    // Wave-uniform core: intersection of both tiles' fully-in-band ranges.
    int cs = (band_lo(n0 + 31) + 3) & ~3;   // align up to K step
    int ce =  band_hi(n0) & ~3;             // align down to K step
    if (cs < kStart) cs = kStart;
    if (ce > kEnd)   ce = kEnd;
    if (ce < cs)     ce = cs;

    const float* xr0 = x + (size_t)(mBase +  0 + col) * N_IN;
    const float* xr1 = x + (size_t)(mBase + 16 + col) * N_IN;
    const float* xr2 = x + (size_t)(mBase + 32 + col) * N_IN;
    const float* xr3 = x + (size_t)(mBase + 48 + col) * N_IN;

    v8f cA0 = {}, cA1 = {}, cA2 = {}, cA3 = {};
    v8f cB0 = {}, cB1 = {}, cB2 = {}, cB3 = {};

    // ---- prologue: band lower edge (<= 8 iterations), branchless selects ----
    for (int kk = kStart; kk < cs; kk += 4) {
        const int k = kk + khalf;
        v2f a0 = *(const v2f*)(xr0 + k);
        v2f a1 = *(const v2f*)(xr1 + k);
        v2f a2 = *(const v2f*)(xr2 + k);
        v2f a3 = *(const v2f*)(xr3 + k);
        v2f bA = load_b_edge(wrowA, k, loA, hiA);
        v2f bB = load_b_edge(wrowB, k, loB, hiB);
        WMMA4(bA, a0, a1, a2, a3, cA0, cA1, cA2, cA3);
        WMMA4(bB, a0, a1, a2, a3, cB0, cB1, cB2, cB3);
    }

    // ---- core: fully in-band, unconditional loads; pointer-bumped + unrolled ----
    {
        const float* p0 = xr0 + cs + khalf;
        const float* p1 = xr1 + cs + khalf;
        const float* p2 = xr2 + cs + khalf;
        const float* p3 = xr3 + cs + khalf;
        const float* wA = wrowA + cs + khalf;
        const float* wB = wrowB + cs + khalf;
        const int steps = (ce - cs) >> 2;
        #pragma unroll 4
        for (int it = 0; it < steps; ++it) {
            v2f a0 = *(const v2f*)p0;
            v2f a1 = *(const v2f*)p1;
            v2f a2 = *(const v2f*)p2;
            v2f a3 = *(const v2f*)p3;
            v2f bA; bA.x = wA[0]; bA.y = wA[1];
            v2f bB; bB.x = wB[0]; bB.y = wB[1];
            WMMA4(bA, a0, a1, a2, a3, cA0, cA1, cA2, cA3);
            WMMA4(bB, a0, a1, a2, a3, cB0, cB1, cB2, cB3);
            p0 += 4; p1 += 4; p2 += 4; p3 += 4; wA += 4; wB += 4;
        }
    }

    // ---- epilogue: band upper edge (<= 8 iterations), branchless selects ----
    for (int kk = ce; kk < kEnd; kk += 4) {
        const int k = kk + khalf;
        v2f a0 = *(const v2f*)(xr0 + k);
        v2f a1 = *(const v2f*)(xr1 + k);
        v2f a2 = *(const v2f*)(xr2 + k);
        v2f a3 = *(const v2f*)(xr3 + k);
        v2f bA = load_b_edge(wrowA, k, loA, hiA);
        v2f bB = load_b_edge(wrowB, k, loB, hiB);
        WMMA4(bA, a0, a1, a2, a3, cA0, cA1, cA2, cA3);
        WMMA4(bB, a0, a1, a2, a3, cB0, cB1, cB2, cB3);
    }

    const float bcA    = bias[oA];
    const float bcB    = bias[oB];
    const int   rowOff = (lane >> 4) << 3;   // 0 or 8: accumulator half-wave M offset

    #pragma unroll
    for (int v = 0; v < 8; ++v) {
        const size_t r = (size_t)(mBase + rowOff + v);
        out[(r +  0) * N_OUT + oA] = cA0[v] + bcA;
        out[(r + 16) * N_OUT + oA] = cA1[v] + bcA;
        out[(r + 32) * N_OUT + oA] = cA2[v] + bcA;
        out[(r + 48) * N_OUT + oA] = cA3[v] + bcA;
        out[(r +  0) * N_OUT + oB] = cB0[v] + bcB;
        out[(r + 16) * N_OUT + oB] = cB1[v] + bcB;
        out[(r + 32) * N_OUT + oB] = cB2[v] + bcB;
        out[(r + 48) * N_OUT + oB] = cB3[v] + bcB;
    }
}

extern "C" void kernel_launch(void* const* d_in, const int* in_sizes, int n_in,
                              void* d_out, int out_size, void* d_ws, size_t ws_size,
                              hipStream_t stream) {
    (void)in_sizes; (void)n_in; (void)out_size; (void)d_ws; (void)ws_size;

    const float* x    = (const float*)d_in[0];
    const float* Wv   = (const float*)d_in[1];
    const float* bias = (const float*)d_in[2];
    float* out = (float*)d_out;

    dim3 grid(N_OUT / 32, M_TOTAL / 512);   // (128, 16): 8 waves/block, 64x32 tile per wave
    band_wmma_f32_kernel<<<grid, 256, 0, stream>>>(x, Wv, bias, out);
}